// TFFastSpeechLengthRegulator_24764781429037
// MI455X (gfx1250) — compile-verified
//
#include <hip/hip_runtime.h>
#include <hip/hip_bf16.h>
#include <stdint.h>

#define BB 32
#define LL 1024
#define HH 384
#define WAVES_PER_BLOCK 8
#define ROW_BYTES (HH * 4)          // 1536 bytes per hidden row

// ---------------------------------------------------------------------------
// Kernel 1: per-batch inclusive cumsum of durations (L=1024 ints per row).
// One 256-thread block per batch row; each thread owns 4 contiguous ints.
// ---------------------------------------------------------------------------
__global__ void lr_cumsum_kernel(const int* __restrict__ dur,
                                 int* __restrict__ cum,
                                 int* __restrict__ total) {
    __shared__ int ssum[256];
    const int b   = blockIdx.x;
    const int tid = threadIdx.x;

    const int4 v = ((const int4*)(dur + (size_t)b * LL))[tid];
    const int s0 = v.x;
    const int s1 = s0 + v.y;
    const int s2 = s1 + v.z;
    const int s3 = s2 + v.w;

    ssum[tid] = s3;
    __syncthreads();

    // Hillis-Steele inclusive scan over the 256 per-thread sums.
    #pragma unroll
    for (int off = 1; off < 256; off <<= 1) {
        const int mine  = ssum[tid];
        const int other = (tid >= off) ? ssum[tid - off] : 0;
        __syncthreads();
        ssum[tid] = mine + other;
        __syncthreads();
    }

    const int excl = (tid == 0) ? 0 : ssum[tid - 1];
    int4 o;
    o.x = s0 + excl; o.y = s1 + excl; o.z = s2 + excl; o.w = s3 + excl;
    ((int4*)(cum + (size_t)b * LL))[tid] = o;

    if (tid == 255) total[b] = o.w;   // cum[L-1] == row total
}

// ---------------------------------------------------------------------------
// Kernel 2: length regulation. One wave32 per output frame t.
// Valid frames are moved with the gfx1250 async LDS DMA path:
//   global_load_async_to_lds_b128  -> s_wait_asynccnt -> global_store_async_from_lds_b128
// (ASYNCcnt-tracked, B128 transactions, no VGPR staging). Masked frames are
// zero-filled with plain float4 stores (d_out is poisoned by the harness).
// ---------------------------------------------------------------------------
__global__ void lr_regulate_kernel(const float* __restrict__ x,
                                   const int* __restrict__ cum,
                                   const int* __restrict__ total,
                                   float* __restrict__ out,
                                   float* __restrict__ mask_out,
                                   int T) {
    __shared__ __align__(16) char smem[WAVES_PER_BLOCK * ROW_BYTES];

    const int lane = threadIdx.x & 31;
    const int wave = threadIdx.x >> 5;
    const int b    = blockIdx.y;
    const int t    = blockIdx.x * WAVES_PER_BLOCK + wave;
    if (t >= T) return;                       // uniform per wave

    const int   tot   = total[b];
    const float maskv = (t < tot) ? 1.0f : 0.0f;
    if (lane == 0) mask_out[(size_t)b * T + t] = maskv;

    float* dst = out + ((size_t)b * T + t) * HH;

    if (t < tot) {
        // searchsorted(cum, t, side='right'): first i with cum[i] > t.
        const int* c = cum + (size_t)b * LL;
        int lo = 0, hi = LL;
        while (lo < hi) {
            const int mid = (lo + hi) >> 1;
            if (c[mid] <= t) lo = mid + 1; else hi = mid;
        }
        const int idx = (lo < (LL - 1)) ? lo : (LL - 1);

        const char* src = (const char*)(x + ((size_t)b * LL + idx) * HH);

        // LDS byte offset of this wave's 1536-byte slab: low 32 bits of the
        // generic shared-memory pointer are the LDS address on gfx12.
        const unsigned ldsbase =
            (unsigned)(uintptr_t)(&smem[wave * ROW_BYTES]) + (unsigned)(lane * 16);

        // 3 x B128 per lane: lanes cover 32*16B = 512B per step, 3 steps = 1536B.
        #pragma unroll
        for (int k = 0; k < 3; ++k) {
            const char*    gsrc   = src + lane * 16 + k * 512;
            const unsigned ldsoff = ldsbase + (unsigned)(k * 512);
            asm volatile("global_load_async_to_lds_b128 %0, %1, off"
                         :: "v"(ldsoff), "v"(gsrc) : "memory");
        }
        asm volatile("s_wait_asynccnt 0x0" ::: "memory");
        #pragma unroll
        for (int k = 0; k < 3; ++k) {
            char*          gdst   = (char*)dst + lane * 16 + k * 512;
            const unsigned ldsoff = ldsbase + (unsigned)(k * 512);
            asm volatile("global_store_async_from_lds_b128 %0, %1, off"
                         :: "v"(gdst), "v"(ldsoff) : "memory");
        }
        // Outstanding async stores drain at the implicit wait-idle of s_endpgm.
    } else {
        const float4 z = make_float4(0.f, 0.f, 0.f, 0.f);
        float4* d4 = (float4*)dst;
        #pragma unroll
        for (int k = 0; k < 3; ++k) d4[lane + 32 * k] = z;
    }
}

// ---------------------------------------------------------------------------
// Host launcher.
//   d_in[0]: encoder_hidden_states (B*L*H f32)
//   d_in[1]: durations_gt          (B*L   i32)
//   d_out  : [ out (B*T*H f32) | mask (B*T f32) ],  T = out_size / (B*(H+1))
//   d_ws   : [ cum (B*L i32) | total (B i32) ]
// ---------------------------------------------------------------------------
extern "C" void kernel_launch(void* const* d_in, const int* in_sizes, int n_in,
                              void* d_out, int out_size, void* d_ws, size_t ws_size,
                              hipStream_t stream) {
    (void)in_sizes; (void)n_in; (void)ws_size;

    const float* x   = (const float*)d_in[0];
    const int*   dur = (const int*)d_in[1];

    const int T = out_size / (BB * (HH + 1));

    int* cum   = (int*)d_ws;
    int* total = cum + (size_t)BB * LL;

    float* out      = (float*)d_out;
    float* mask_out = out + (size_t)BB * T * HH;

    lr_cumsum_kernel<<<BB, 256, 0, stream>>>(dur, cum, total);

    dim3 grid((T + WAVES_PER_BLOCK - 1) / WAVES_PER_BLOCK, BB);
    lr_regulate_kernel<<<grid, 256, 0, stream>>>(x, cum, total, out, mask_out, T);
}